// RoiDeform_29343216566875
// MI455X (gfx1250) — compile-verified
//
#include <hip/hip_runtime.h>
#include <hip/hip_bf16.h>
#include <math.h>

// RoiDeform: B=2, H=W=1600, C=3, k=4, crop_size=384.
// Output (2,16,384,384,3) f32 = 56.6 MB; every output row within a bin is a
// copy of one 384-sample bilinear row -> stage row in LDS, replicate with
// CDNA5 async LDS->global B128 stores (ASYNCcnt).

#define IMG_H   1600
#define IMG_W   1600
#define CS      384
#define KK      4
#define PAD     32                 // (1600 - 384*4)/2
#define ROW_F   (CS * 3)           // 1152 floats per output row
#define ROW_B   (ROW_F * 4)        // 4608 bytes per output row
#define ROWS_PER_BLK 16
#define CHUNKS  (CS / ROWS_PER_BLK)  // 24
#define THREADS 256
#define XFERS   (ROWS_PER_BLK * (ROW_B / 16))  // 16 * 288 = 4608 float4 moves

__global__ __launch_bounds__(THREADS)
void roi_deform_broadcast(const float* __restrict__ img,
                          const float* __restrict__ delta_p,
                          float* __restrict__ out) {
    __shared__ __align__(16) float rowbuf[ROW_F];

    const int tid   = threadIdx.x;
    const int bid   = blockIdx.x;
    const int chunk = bid % CHUNKS;          // which 16-row slice of the bin
    const int binId = bid / CHUNKS;          // 0..31  (t*16 + bin)
    const int t     = binId >> 4;
    const int bin   = binId & 15;
    const int bi    = bin >> 2;              // i (row bin)
    const int bj    = bin & 3;               // j (col bin)

    const int dpb = ((t * KK + bi) * KK + bj) * 2;
    const float dp0 = delta_p[dpb + 0];
    const float dp1 = delta_p[dpb + 1];
    const float rbase = (float)(PAD + bi * CS);
    const float cbase = (float)(PAD + bj * CS);

    // ---- Phase 1: compute the bin's unique 384-sample row into LDS ----
    for (int s = tid; s < CS; s += THREADS) {
        // (pad + bin*cs + s) is an exact small integer; one rounding on +dp,
        // matching the reference's float add order.
        const float rv = (rbase + (float)s) + dp0;
        const float cv = (cbase + (float)s) + dp1;
        const float rfl = floorf(rv), cfl = floorf(cv);
        const float rf = rv - rfl, cf = cv - cfl;

        float a0 = 0.0f, a1 = 0.0f, a2 = 0.0f;
#pragma unroll
        for (int dr = 0; dr < 2; ++dr) {
            const float ri = rfl + (float)dr;
            const float wr = dr ? rf : (1.0f - rf);
            const bool rvalid = (ri >= 0.0f) && (ri < (float)IMG_H);
            const int  rii = (int)fminf(fmaxf(ri, 0.0f), (float)(IMG_H - 1));
#pragma unroll
            for (int dc = 0; dc < 2; ++dc) {
                const float ci = cfl + (float)dc;
                const float wc = dc ? cf : (1.0f - cf);
                const bool valid = rvalid && (ci >= 0.0f) && (ci < (float)IMG_W);
                if (valid) {
                    const int cii = (int)ci;   // in [0,1599] when valid
                    const float w = wr * wc;
                    const float* p = img +
                        ((size_t)((t * IMG_H + rii) * IMG_W + cii)) * 3;
                    a0 += w * p[0];
                    a1 += w * p[1];
                    a2 += w * p[2];
                }
            }
        }
        rowbuf[s * 3 + 0] = a0;
        rowbuf[s * 3 + 1] = a1;
        rowbuf[s * 3 + 2] = a2;
    }
    __syncthreads();

    // ---- Phase 2: replicate the row to 16 output rows via async LDS->global
    // B128 stores (per-lane 16B, GV mode, tracked by ASYNCcnt). ----
    const unsigned long long gbase =
        (unsigned long long)(size_t)out +
        (unsigned long long)((binId * CS) + chunk * ROWS_PER_BLK) *
            (unsigned long long)ROW_B;
    // Low 32 bits of the generic address of a __shared__ object are the
    // wave-relative LDS byte offset (aperture lives in the high bits).
    const unsigned lbase = (unsigned)(size_t)(&rowbuf[0]);

    // XFERS = 4608 = 18 * THREADS exactly -> no divergence, EXEC all-ones.
    for (int idx = tid; idx < XFERS; idx += THREADS) {
        const int r  = idx / (ROW_B / 16);
        const int s4 = idx - r * (ROW_B / 16);
        const unsigned long long g =
            gbase + (unsigned long long)r * (unsigned long long)ROW_B +
            (unsigned long long)(s4 * 16);
        const unsigned l = lbase + (unsigned)(s4 * 16);
        asm volatile("global_store_async_from_lds_b128 %0, %1, off"
                     :: "v"(g), "v"(l)
                     : "memory");
    }

#if __has_builtin(__builtin_amdgcn_s_wait_asynccnt)
    __builtin_amdgcn_s_wait_asynccnt(0);
#else
    asm volatile("s_wait_asynccnt 0x0" ::: "memory");
#endif
}

extern "C" void kernel_launch(void* const* d_in, const int* in_sizes, int n_in,
                              void* d_out, int out_size, void* d_ws, size_t ws_size,
                              hipStream_t stream) {
    (void)in_sizes; (void)n_in; (void)out_size; (void)d_ws; (void)ws_size;
    const float* img = (const float*)d_in[0];   // (2,1600,1600,3) f32
    const float* dp  = (const float*)d_in[1];   // (2,4,4,2) f32
    float* out = (float*)d_out;                 // (2,16,384,384,3) f32

    dim3 grid(2 * KK * KK * CHUNKS);            // 32 bins * 24 chunks = 768
    dim3 block(THREADS);
    hipLaunchKernelGGL(roi_deform_broadcast, grid, block, 0, stream,
                       img, dp, out);
}